// VariancePenalization_11639361372258
// MI455X (gfx1250) — compile-verified
//
#include <hip/hip_runtime.h>
#include <math.h>

// ---------------------------------------------------------------------------
// Distributionally-robust best-response weights on MI455X (gfx1250, wave32).
//
// f(eta) = m*Q/S^2 - 1 - lambda,  S = sum relu(v-eta), Q = sum relu(v-eta)^2.
// Instead of one 128MB pass per bisection step (~30-60 passes), build a
// (count, sum, sumsq) histogram once, run the whole bracket-expansion +
// bisection on 48KB of suffix sums in LDS, refine with a second histogram
// over the localized bucket window, then one exact pass for dot(p, v).
// 4 streaming passes total; v is L2-resident (192MB L2) after the first.
//
// WMMA F32 16x16x4 with an all-ones A matrix is the summation engine for the
// streaming reductions (D[m][n] = sum_k B[k][n]; total = sum(D)/16).
// All streaming loads are 128-bit (global_load_b128) to minimize VMEM issue
// per byte at 23.3 TB/s; prefetch pulls into the near cache for reread passes.
// ---------------------------------------------------------------------------

typedef float v2f __attribute__((ext_vector_type(2)));
typedef float v8f __attribute__((ext_vector_type(8)));

#define NBINS 4096
#define HDRW  16
#define SOLVE_THREADS 1024
#define CHUNK (NBINS / SOLVE_THREADS)

#define LMBDA_F 1.0f
#define TOL_F   1e-4f
#define MAX_ITER 1000
#define MIN_REL_DIFF 1e-5f

// ws 32-bit word layout:
//  [0] ord(max)  [1] ord(min)  [2] sum(v)  [3] unused
//  [4] lo1  [5] w1  (level-0 hist params)
//  [6] lo2  [7] w2  (level-1 hist params; 0 => skip)
//  [8] baseC [9] baseSv [10] baseSq  (suffix above level-1 window)
//  [11] done flag  [12] eta*  [13] num  [14] den  [15] uniform flag
//  [16 ..]            hist0: cnt[NBINS], sv[NBINS], sq[NBINS]
//  [16+3*NBINS ..]    hist1: cnt[NBINS], sv[NBINS], sq[NBINS]

__device__ __forceinline__ unsigned f2ord(float f) {
  unsigned u = __float_as_uint(f);
  return (u & 0x80000000u) ? ~u : (u | 0x80000000u);
}
__device__ __forceinline__ float ord2f(unsigned u) {
  return (u & 0x80000000u) ? __uint_as_float(u & 0x7fffffffu)
                           : __uint_as_float(~u);
}

__device__ __forceinline__ float waveRedAdd(float x) {
#pragma unroll
  for (int o = 16; o > 0; o >>= 1) x += __shfl_xor(x, o, 32);
  return x;
}
__device__ __forceinline__ float waveRedMax(float x) {
#pragma unroll
  for (int o = 16; o > 0; o >>= 1) x = fmaxf(x, __shfl_xor(x, o, 32));
  return x;
}
__device__ __forceinline__ float waveRedMin(float x) {
#pragma unroll
  for (int o = 16; o > 0; o >>= 1) x = fminf(x, __shfl_xor(x, o, 32));
  return x;
}

// D += ones(16x4) * B(4x16): every row of D gets B's column sums, so
// sum over all D entries counts every B element exactly 16 times.
__device__ __forceinline__ v8f wmma_rowsum(v2f b, v8f c) {
  v2f ones;
  ones.x = 1.0f;
  ones.y = 1.0f;
  return __builtin_amdgcn_wmma_f32_16x16x4_f32(
      /*neg_a=*/false, ones, /*neg_b=*/false, b,
      /*c_mod=*/(short)0, c, /*reuse_a=*/false, /*reuse_b=*/false);
}

__global__ void k_init(unsigned* wsu, int words) {
  int i = blockIdx.x * blockDim.x + threadIdx.x;
  if (i >= words) return;
  unsigned val = 0u;
  if (i == 1) val = 0xffffffffu;  // atomicMin init for ord(min)
  wsu[i] = val;
}

// Pass 1: global max, min, sum(v). Sum rides the WMMA pipe; 128-bit loads.
__global__ __launch_bounds__(256) void k_pass1(const float* __restrict__ v,
                                               int n, float* wsf,
                                               unsigned* wsu) {
  const int total = gridDim.x * blockDim.x;
  const int gid = blockIdx.x * blockDim.x + threadIdx.x;
  const int n4 = n >> 2;
  const int niter = n4 / total;  // uniform scalar trip count: EXEC all-ones
  const float4* __restrict__ v4 = (const float4*)v;

  float mx = -3.0e38f, mn = 3.0e38f, tail = 0.0f;
  v8f acc = {};
  for (int it = 0; it < niter; ++it) {
    int idx = gid + it * total;
    float4 x = v4[idx];
    __builtin_prefetch(v4 + idx + 8 * total, 0, 3);
    mx = fmaxf(mx, fmaxf(fmaxf(x.x, x.y), fmaxf(x.z, x.w)));
    mn = fminf(mn, fminf(fminf(x.x, x.y), fminf(x.z, x.w)));
    v2f b0, b1;
    b0.x = x.x; b0.y = x.y;
    b1.x = x.z; b1.y = x.w;
    acc = wmma_rowsum(b0, acc);
    acc = wmma_rowsum(b1, acc);
  }
  for (int idx = gid + niter * total; idx < n4; idx += total) {  // no WMMA here
    float4 x = v4[idx];
    mx = fmaxf(mx, fmaxf(fmaxf(x.x, x.y), fmaxf(x.z, x.w)));
    mn = fminf(mn, fminf(fminf(x.x, x.y), fminf(x.z, x.w)));
    tail += (x.x + x.y) + (x.z + x.w);
  }
  if (gid == 0) {
    for (int idx = n4 << 2; idx < n; ++idx) {  // n % 4 remainder
      float x = v[idx];
      mx = fmaxf(mx, x);
      mn = fminf(mn, x);
      tail += x;
    }
  }
  float p = (acc[0] + acc[1] + acc[2] + acc[3] + acc[4] + acc[5] + acc[6] +
             acc[7]) * (1.0f / 16.0f) + tail;
  p = waveRedAdd(p);
  mx = waveRedMax(mx);
  mn = waveRedMin(mn);
  if ((threadIdx.x & 31) == 0) {
    atomicAdd(&wsf[2], p);
    atomicMax(&wsu[0], f2ord(mx));
    atomicMin(&wsu[1], f2ord(mn));
  }
}

__global__ void k_prep(float* wsf, unsigned* wsu) {
  float vmax = ord2f(wsu[0]);
  float vmin = ord2f(wsu[1]);
  float range = vmax - vmin;
  if (!(range > 0.0f)) range = 1.0f;
  wsf[4] = vmin;
  wsf[5] = range * (1.000001f / (float)NBINS);  // inflate so vmax lands inside
}

// Histogram pass: LDS (cnt, sum, sumsq) per bucket, flushed with global atomics.
__global__ __launch_bounds__(256) void k_hist(const float* __restrict__ v,
                                              int n, float* wsf, unsigned* wsu,
                                              int level) {
  __shared__ unsigned s_cnt[NBINS];
  __shared__ float s_sv[NBINS];
  __shared__ float s_sq[NBINS];
  for (int i = threadIdx.x; i < NBINS; i += blockDim.x) {
    s_cnt[i] = 0u;
    s_sv[i] = 0.0f;
    s_sq[i] = 0.0f;
  }
  const float lo = wsf[4 + 2 * level];
  const float w = wsf[5 + 2 * level];
  __syncthreads();

  if (w > 0.0f) {
    const float invw = 1.0f / w;
    const int total = gridDim.x * blockDim.x;
    const int gid = blockIdx.x * blockDim.x + threadIdx.x;
    const int n4 = n >> 2;
    const float4* __restrict__ v4 = (const float4*)v;
    for (int idx = gid; idx < n4; idx += total) {
      float4 x = v4[idx];
      __builtin_prefetch(v4 + idx + 8 * total, 0, 3);
      float e[4] = {x.x, x.y, x.z, x.w};
#pragma unroll
      for (int k = 0; k < 4; ++k) {
        float xv = e[k];
        int b = (int)floorf((xv - lo) * invw);
        bool in;
        if (level == 0) {
          b = max(0, min(b, NBINS - 1));
          in = true;
        } else {
          in = (b >= 0) && (b < NBINS);
        }
        if (in) {
          atomicAdd(&s_cnt[b], 1u);   // ds_add_u32
          atomicAdd(&s_sv[b], xv);    // ds_add_f32
          atomicAdd(&s_sq[b], xv * xv);
        }
      }
    }
    if (gid == 0) {
      for (int idx = n4 << 2; idx < n; ++idx) {  // n % 4 remainder
        float xv = v[idx];
        int b = (int)floorf((xv - lo) * invw);
        bool in;
        if (level == 0) {
          b = max(0, min(b, NBINS - 1));
          in = true;
        } else {
          in = (b >= 0) && (b < NBINS);
        }
        if (in) {
          atomicAdd(&s_cnt[b], 1u);
          atomicAdd(&s_sv[b], xv);
          atomicAdd(&s_sq[b], xv * xv);
        }
      }
    }
  }
  __syncthreads();
  const int base = HDRW + level * 3 * NBINS;
  for (int i = threadIdx.x; i < NBINS; i += blockDim.x) {
    if (s_cnt[i]) {
      atomicAdd(&wsu[base + i], s_cnt[i]);
      atomicAdd(&wsf[base + NBINS + i], s_sv[i]);
      atomicAdd(&wsf[base + 2 * NBINS + i], s_sq[i]);
    }
  }
}

// Suffix-scan the histogram in LDS, then run bracket expansion + bisection
// entirely on-chip (thread 0). level 0 also localizes the refinement window.
__global__ __launch_bounds__(SOLVE_THREADS) void k_solve(int n, float* wsf,
                                                         unsigned* wsu,
                                                         int level) {
  __shared__ float sC[NBINS], sSv[NBINS], sSq[NBINS];
  __shared__ float pC[SOLVE_THREADS], pV[SOLVE_THREADS], pQ[SOLVE_THREADS];

  const int t = threadIdx.x;
  const int base = HDRW + level * 3 * NBINS;
  float lc[CHUNK], lv[CHUNK], lq[CHUNK];
  float cc = 0.0f, cs = 0.0f, cq = 0.0f;
#pragma unroll
  for (int i = 0; i < CHUNK; ++i) {
    int b = t * CHUNK + i;
    lc[i] = (float)wsu[base + b];
    lv[i] = wsf[base + NBINS + b];
    lq[i] = wsf[base + 2 * NBINS + b];
    cc += lc[i];
    cs += lv[i];
    cq += lq[i];
  }
  pC[t] = cc;
  pV[t] = cs;
  pQ[t] = cq;
  __syncthreads();
  for (int off = 1; off < SOLVE_THREADS; off <<= 1) {  // inclusive suffix scan
    float aC = (t + off < SOLVE_THREADS) ? pC[t + off] : 0.0f;
    float aV = (t + off < SOLVE_THREADS) ? pV[t + off] : 0.0f;
    float aQ = (t + off < SOLVE_THREADS) ? pQ[t + off] : 0.0f;
    __syncthreads();
    pC[t] += aC;
    pV[t] += aV;
    pQ[t] += aQ;
    __syncthreads();
  }
  float eC = (t < SOLVE_THREADS - 1) ? pC[t + 1] : 0.0f;
  float eV = (t < SOLVE_THREADS - 1) ? pV[t + 1] : 0.0f;
  float eQ = (t < SOLVE_THREADS - 1) ? pQ[t + 1] : 0.0f;
#pragma unroll
  for (int i = CHUNK - 1; i >= 0; --i) {
    eC += lc[i];
    eV += lv[i];
    eQ += lq[i];
    sC[t * CHUNK + i] = eC;
    sSv[t * CHUNK + i] = eV;
    sSq[t * CHUNK + i] = eQ;
  }
  __syncthreads();

  if (t == 0) {
    const float m = (float)n;
    if (level == 0) {
      float vmax = ord2f(wsu[0]);
      float vmin = ord2f(wsu[1]);
      float rel = (vmax - vmin) / vmax;  // NaN -> not uniform (matches ref)
      bool uniform = (rel <= MIN_REL_DIFF);
      wsf[15] = uniform ? 1.0f : 0.0f;
      if (uniform) {
        wsu[11] = 1u;
        wsf[12] = vmax;
      } else {
        const float lo1 = wsf[4], w1 = wsf[5];
        const float invw = 1.0f / w1;
        auto evalF = [&](float eta) -> float {
          float Cc, Sv, Sq;
          if (eta < lo1) {  // exact region: all elements included
            Cc = sC[0];
            Sv = sSv[0];
            Sq = sSq[0];
          } else {
            int b = (int)floorf((eta - lo1) * invw) + 1;
            if (b < 0) b = 0;
            if (b >= NBINS) {
              Cc = 0.0f; Sv = 0.0f; Sq = 0.0f;
            } else {
              Cc = sC[b]; Sv = sSv[b]; Sq = sSq[b];
            }
          }
          float S = Sv - eta * Cc;
          if (!(S > 0.0f)) return 3.0e38f;
          float Q = Sq - 2.0f * eta * Sv + eta * eta * Cc;
          return m * Q / (S * S) - 1.0f - LMBDA_F;
        };
        float eta_min = -(1.0f / (sqrtf(2.0f * LMBDA_F + 1.0f) - 1.0f)) * vmax;
        float eta_max = vmax;
        float flo = evalF(eta_min), fhi = evalF(eta_max);
        for (int g = 0; g < 200 && ((flo > 0.0f) || (fhi < 0.0f)); ++g) {
          float len = eta_max - eta_min;
          float hi2 = (flo > 0.0f) ? eta_min : eta_max;
          float lo2x = (flo > 0.0f) ? (eta_min - 2.0f * len) : eta_min;
          float lo3 = (fhi < 0.0f) ? hi2 : lo2x;
          float hi3 = (fhi < 0.0f) ? (hi2 + 2.0f * len) : hi2;
          eta_min = lo3;
          eta_max = hi3;
          flo = evalF(eta_min);
          fhi = evalF(eta_max);
        }
        float lo = eta_min, hi = eta_max;
        float eta = 0.5f * (lo + hi), val = evalF(eta);
        for (int i = 1; (i < MAX_ITER) && (fabsf(val) > TOL_F); ++i) {
          if (val > 0.0f) hi = eta;
          else if (val < 0.0f) lo = eta;
          eta = 0.5f * (lo + hi);
          val = evalF(eta);
        }
        if (val > 0.0f) hi = eta;
        else if (val < 0.0f) lo = eta;
        float etaStar = (fabsf(val) <= TOL_F) ? eta : 0.5f * (lo + hi);
        wsf[12] = etaStar;
        if (etaStar < lo1) {
          wsu[11] = 1u;  // root in exact region, no refinement needed
        } else {
          int bs = (int)floorf((etaStar - lo1) * invw);
          int blo = bs - 2; if (blo < 0) blo = 0;
          int bhi = bs + 3; if (bhi > NBINS) bhi = NBINS;
          wsf[6] = lo1 + (float)blo * w1;
          wsf[7] = ((float)(bhi - blo)) * w1 / (float)NBINS;
          float bc = 0.0f, bv = 0.0f, bq = 0.0f;
          if (bhi < NBINS) { bc = sC[bhi]; bv = sSv[bhi]; bq = sSq[bhi]; }
          wsf[8] = bc; wsf[9] = bv; wsf[10] = bq;
          wsu[11] = 0u;
        }
      }
    } else if (wsu[11] == 0u) {
      const float lo2 = wsf[6], w2 = wsf[7];
      const float invw2 = 1.0f / w2;
      const float bc = wsf[8], bv = wsf[9], bq = wsf[10];
      auto eval2 = [&](float eta) -> float {
        int b = (int)floorf((eta - lo2) * invw2) + 1;
        if (b < 0) b = 0;
        float Cc = bc, Sv = bv, Sq = bq;
        if (b < NBINS) { Cc += sC[b]; Sv += sSv[b]; Sq += sSq[b]; }
        float S = Sv - eta * Cc;
        if (!(S > 0.0f)) return 3.0e38f;
        float Q = Sq - 2.0f * eta * Sv + eta * eta * Cc;
        return m * Q / (S * S) - 1.0f - LMBDA_F;
      };
      float lo = lo2, hi = lo2 + (float)NBINS * w2;
      float flo = eval2(lo), fhi = eval2(hi);
      float etaStar;
      if (flo > 0.0f) etaStar = lo;
      else if (fhi < 0.0f) etaStar = hi;
      else {
        float eta = 0.5f * (lo + hi), val = eval2(eta);
        for (int i = 1; (i < 200) && (fabsf(val) > 0.25f * TOL_F); ++i) {
          if (val > 0.0f) hi = eta;
          else if (val < 0.0f) lo = eta;
          eta = 0.5f * (lo + hi);
          val = eval2(eta);
        }
        etaStar = eta;
      }
      wsf[12] = etaStar;
    }
  }
}

// Exact final pass: num = sum relu(v-eta)*v, den = sum relu(v-eta), via WMMA.
__global__ __launch_bounds__(256) void k_final(const float* __restrict__ v,
                                               int n, float* wsf) {
  const float eta = wsf[12];
  const int total = gridDim.x * blockDim.x;
  const int gid = blockIdx.x * blockDim.x + threadIdx.x;
  const int n4 = n >> 2;
  const int niter = n4 / total;
  const float4* __restrict__ v4 = (const float4*)v;

  v8f cn = {}, cd = {};
  float tn = 0.0f, td = 0.0f;
  for (int it = 0; it < niter; ++it) {
    int idx = gid + it * total;
    float4 x = v4[idx];
    __builtin_prefetch(v4 + idx + 8 * total, 0, 3);
    float rx = fmaxf(x.x - eta, 0.0f);
    float ry = fmaxf(x.y - eta, 0.0f);
    float rz = fmaxf(x.z - eta, 0.0f);
    float rw = fmaxf(x.w - eta, 0.0f);
    v2f bn0, bn1, bd0, bd1;
    bn0.x = rx * x.x; bn0.y = ry * x.y;
    bn1.x = rz * x.z; bn1.y = rw * x.w;
    bd0.x = rx;       bd0.y = ry;
    bd1.x = rz;       bd1.y = rw;
    cn = wmma_rowsum(bn0, cn);
    cn = wmma_rowsum(bn1, cn);
    cd = wmma_rowsum(bd0, cd);
    cd = wmma_rowsum(bd1, cd);
  }
  for (int idx = gid + niter * total; idx < n4; idx += total) {  // no WMMA here
    float4 x = v4[idx];
    float rx = fmaxf(x.x - eta, 0.0f);
    float ry = fmaxf(x.y - eta, 0.0f);
    float rz = fmaxf(x.z - eta, 0.0f);
    float rw = fmaxf(x.w - eta, 0.0f);
    tn += (rx * x.x + ry * x.y) + (rz * x.z + rw * x.w);
    td += (rx + ry) + (rz + rw);
  }
  if (gid == 0) {
    for (int idx = n4 << 2; idx < n; ++idx) {  // n % 4 remainder
      float x = v[idx];
      float r = fmaxf(x - eta, 0.0f);
      tn += r * x;
      td += r;
    }
  }
  float pn = (cn[0] + cn[1] + cn[2] + cn[3] + cn[4] + cn[5] + cn[6] + cn[7]) *
                 (1.0f / 16.0f) + tn;
  float pd = (cd[0] + cd[1] + cd[2] + cd[3] + cd[4] + cd[5] + cd[6] + cd[7]) *
                 (1.0f / 16.0f) + td;
  pn = waveRedAdd(pn);
  pd = waveRedAdd(pd);
  if ((threadIdx.x & 31) == 0) {
    atomicAdd(&wsf[13], pn);
    atomicAdd(&wsf[14], pd);
  }
}

__global__ void k_out(const float* wsf, float* out, int n) {
  float res;
  if (wsf[15] != 0.0f) {
    res = wsf[2] / (float)n;  // uniform weights -> mean(v)
  } else {
    float den = wsf[14];
    res = (den != 0.0f) ? (wsf[13] / den) : (wsf[2] / (float)n);
  }
  out[0] = res;
}

extern "C" void kernel_launch(void* const* d_in, const int* in_sizes, int n_in,
                              void* d_out, int out_size, void* d_ws,
                              size_t ws_size, hipStream_t stream) {
  (void)n_in; (void)out_size; (void)ws_size;
  const float* v = (const float*)d_in[0];
  const int n = in_sizes[0];
  float* wsf = (float*)d_ws;
  unsigned* wsu = (unsigned*)d_ws;
  float* out = (float*)d_out;

  const int W = HDRW + 2 * 3 * NBINS;
  k_init<<<(W + 255) / 256, 256, 0, stream>>>(wsu, W);
  k_pass1<<<1024, 256, 0, stream>>>(v, n, wsf, wsu);
  k_prep<<<1, 1, 0, stream>>>(wsf, wsu);
  k_hist<<<1024, 256, 0, stream>>>(v, n, wsf, wsu, 0);
  k_solve<<<1, SOLVE_THREADS, 0, stream>>>(n, wsf, wsu, 0);
  k_hist<<<1024, 256, 0, stream>>>(v, n, wsf, wsu, 1);
  k_solve<<<1, SOLVE_THREADS, 0, stream>>>(n, wsf, wsu, 1);
  k_final<<<1024, 256, 0, stream>>>(v, n, wsf);
  k_out<<<1, 1, 0, stream>>>(wsf, out, n);
}